// SGCModel_11622181503743
// MI455X (gfx1250) — compile-verified
//
#include <hip/hip_runtime.h>
#include <hip/hip_bf16.h>

#define N_NODES   50000
#define D_FEAT    128
#define HIDDEN    128
#define N_CLASSES 40

typedef __attribute__((ext_vector_type(2))) float v2f;
typedef __attribute__((ext_vector_type(8))) float v8f;

// ---------- GCN normalization ----------
__global__ void deg_init_kernel(float* deg, int n) {
    int i = blockIdx.x * blockDim.x + threadIdx.x;
    if (i < n) deg[i] = 1.0f;  // self-loop contributes 1 to every node's degree
}

__global__ void deg_edges_kernel(const int* __restrict__ dst, float* deg, int nE) {
    int e = blockIdx.x * blockDim.x + threadIdx.x;
    if (e < nE) atomicAdd(&deg[dst[e]], 1.0f);
}

__global__ void dinv_kernel(float* deg, int n) {
    int i = blockIdx.x * blockDim.x + threadIdx.x;
    if (i < n) {
        float d = deg[i];
        deg[i] = (d > 0.0f) ? rsqrtf(d) : 0.0f;  // in-place: deg -> dinv
    }
}

// ---------- Propagation hop ----------
// Self-loop term initializes the output: hout = dinv[i]^2 * hin  (norm for i->i)
__global__ void selfloop_kernel(const float* __restrict__ hin,
                                const float* __restrict__ dinv,
                                float* __restrict__ hout, int n4) {
    int i = blockIdx.x * blockDim.x + threadIdx.x;  // float4 index
    if (i < n4) {
        int row = i >> 5;                    // 32 float4 per 128-float row
        float dv = dinv[row];
        float s = dv * dv;
        float4 v = ((const float4*)hin)[i];
        v.x *= s; v.y *= s; v.z *= s; v.w *= s;
        ((float4*)hout)[i] = v;
    }
}

// One wave32 per edge: 32 lanes x float4 = 128 features.
__global__ void hop_edges_kernel(const int* __restrict__ src,
                                 const int* __restrict__ dst,
                                 const float* __restrict__ dinv,
                                 const float* __restrict__ hin,
                                 float* __restrict__ hout, int nE) {
    int gid  = blockIdx.x * blockDim.x + threadIdx.x;
    int e    = gid >> 5;
    int lane = threadIdx.x & 31;
    if (e >= nE) return;
    int s = src[e];
    int d = dst[e];
    float norm = dinv[s] * dinv[d];
    float4 v = ((const float4*)(hin + (size_t)s * D_FEAT))[lane];
    float* dp = hout + (size_t)d * D_FEAT + lane * 4;
    atomicAdd(dp + 0, norm * v.x);
    atomicAdd(dp + 1, norm * v.y);
    atomicAdd(dp + 2, norm * v.z);
    atomicAdd(dp + 3, norm * v.w);
}

// ---------- GEMM 1: C[50000x128] = H[50000x128] @ W[128x128] + b ----------
// 8 waves/block, one 16x16 WMMA tile per wave (wave -> n-tile, block -> m-tile).
__global__ __launch_bounds__(256) void gemm_sgc_kernel(const float* __restrict__ H,
                                                       const float* __restrict__ W,
                                                       const float* __restrict__ bias,
                                                       float* __restrict__ C) {
    const int wave = threadIdx.x >> 5;
    const int lane = threadIdx.x & 31;
    const int m0 = blockIdx.x * 16;
    const int n0 = wave * 16;
    const int lm = lane & 15;            // M index (A) / N index (B,C)
    const int kh = (lane >> 4) * 2;      // K sub-offset: 0 or 2

    v8f c = {};
    const float* Arow = H + (size_t)(m0 + lm) * HIDDEN;
#pragma unroll 4
    for (int k0 = 0; k0 < HIDDEN; k0 += 4) {
        v2f a, b;
        a.x = Arow[k0 + kh];
        a.y = Arow[k0 + kh + 1];
        b.x = W[(size_t)(k0 + kh) * HIDDEN + n0 + lm];
        b.y = W[(size_t)(k0 + kh + 1) * HIDDEN + n0 + lm];
        c = __builtin_amdgcn_wmma_f32_16x16x4_f32(false, a, false, b,
                                                  (short)0, c, false, false);
    }
    const float bv = bias[n0 + lm];
    const int rh = (lane >> 4) * 8;      // row half: M offset 0 or 8
#pragma unroll
    for (int i = 0; i < 8; ++i)
        C[(size_t)(m0 + i + rh) * HIDDEN + n0 + lm] = c[i] + bv;
}

// ---------- GEMM 2: Out[50000x40] = H[50000x128] @ W[128x40] + b ----------
// 3 waves/block cover N padded to 48; column guard via select (EXEC stays all-1s
// through the WMMA loop).
__global__ __launch_bounds__(96) void gemm_cls_kernel(const float* __restrict__ H,
                                                      const float* __restrict__ W,
                                                      const float* __restrict__ bias,
                                                      float* __restrict__ Out) {
    const int wave = threadIdx.x >> 5;
    const int lane = threadIdx.x & 31;
    const int m0 = blockIdx.x * 16;
    const int n0 = wave * 16;
    const int lm = lane & 15;
    const int kh = (lane >> 4) * 2;
    const int n  = n0 + lm;
    const bool nv = (n < N_CLASSES);

    v8f c = {};
    const float* Arow = H + (size_t)(m0 + lm) * HIDDEN;
#pragma unroll 4
    for (int k0 = 0; k0 < HIDDEN; k0 += 4) {
        v2f a, b;
        a.x = Arow[k0 + kh];
        a.y = Arow[k0 + kh + 1];
        b.x = nv ? W[(size_t)(k0 + kh) * N_CLASSES + n] : 0.0f;
        b.y = nv ? W[(size_t)(k0 + kh + 1) * N_CLASSES + n] : 0.0f;
        c = __builtin_amdgcn_wmma_f32_16x16x4_f32(false, a, false, b,
                                                  (short)0, c, false, false);
    }
    if (nv) {
        const float bv = bias[n];
        const int rh = (lane >> 4) * 8;
#pragma unroll
        for (int i = 0; i < 8; ++i)
            Out[(size_t)(m0 + i + rh) * N_CLASSES + n] = c[i] + bv;
    }
}

extern "C" void kernel_launch(void* const* d_in, const int* in_sizes, int n_in,
                              void* d_out, int out_size, void* d_ws, size_t ws_size,
                              hipStream_t stream) {
    const float* x    = (const float*)d_in[0];
    const int*   ei   = (const int*)d_in[1];    // [2, E] flattened: src row then dst row
    const float* Wsgc = (const float*)d_in[2];
    const float* bsgc = (const float*)d_in[3];
    const float* Wcls = (const float*)d_in[4];
    const float* bcls = (const float*)d_in[5];
    float* out = (float*)d_out;

    const int nE = in_sizes[1] / 2;
    const int* src = ei;
    const int* dst = ei + nE;

    char* ws = (char*)d_ws;
    float* dinv = (float*)ws;                                   // 50000 floats
    float* hA   = (float*)(ws + 262144);                        // 25.6 MB
    float* hB   = hA + (size_t)N_NODES * D_FEAT;                // 25.6 MB

    const int n4 = N_NODES * (D_FEAT / 4);
    const int mTiles = N_NODES / 16;  // 3125

    // GCN norm
    deg_init_kernel<<<(N_NODES + 255) / 256, 256, 0, stream>>>(dinv, N_NODES);
    deg_edges_kernel<<<(nE + 255) / 256, 256, 0, stream>>>(dst, dinv, nE);
    dinv_kernel<<<(N_NODES + 255) / 256, 256, 0, stream>>>(dinv, N_NODES);

    // Hop 1: x -> hA
    selfloop_kernel<<<(n4 + 255) / 256, 256, 0, stream>>>(x, dinv, hA, n4);
    hop_edges_kernel<<<(nE + 7) / 8, 256, 0, stream>>>(src, dst, dinv, x, hA, nE);

    // Hop 2: hA -> hB
    selfloop_kernel<<<(n4 + 255) / 256, 256, 0, stream>>>(hA, dinv, hB, n4);
    hop_edges_kernel<<<(nE + 7) / 8, 256, 0, stream>>>(src, dst, dinv, hA, hB, nE);

    // SGConv linear: hA = hB @ Wsgc + bsgc   (WMMA f32)
    gemm_sgc_kernel<<<mTiles, 256, 0, stream>>>(hB, Wsgc, bsgc, hA);

    // Classifier: out = hA @ Wcls + bcls     (WMMA f32)
    gemm_cls_kernel<<<mTiles, 96, 0, stream>>>(hA, Wcls, bcls, out);
}